// TorchSimOrderParameters_82068235092606
// MI455X (gfx1250) — compile-verified
//
#include <hip/hip_runtime.h>
#include <hip/hip_bf16.h>

typedef __attribute__((ext_vector_type(2))) float v2f;
typedef __attribute__((ext_vector_type(8))) float v8f;

// tet constants
#define THETA_T   1.9106332362490186f   // 0.6081734479693927 * pi
#define INV_SIGMA 4.7746482927568605f   // 1 / (12 deg in rad)

// Steinhardt normalizations nlm = sqrt((2l+1)/(4pi) * (l-m)!/(l+m)!)
#define N20 0.6307831305050401f
#define N21 0.2575161346821264f
#define N22 0.1287580673410632f
#define N40 0.8462843753216345f
#define N41 0.18923493915151202f
#define N42 0.04460301029170167f
#define N43 0.011920680425156177f
#define N44 0.004214597094385378f
#define N60 1.0171072362820548f
#define N61 0.15694305382561464f
#define N62 0.024815077351764973f
#define N63 0.004135812590201913f
#define N64 7.550926197134423e-4f
#define N65 1.6098630382781359e-4f
#define N66 4.647274205848904e-5f

__device__ __forceinline__ float wred(float v) {
#pragma unroll
    for (int s = 16; s > 0; s >>= 1) v += __shfl_xor(v, s, 32);
    return v;
}

// One (re,im) spherical-harmonic term: wave-reduce P*cos(m phi)*mask and
// P*sin(m phi)*mask, accumulate w*(re^2+im^2).
__device__ __forceinline__ void qterm(float P, float mphi, float mk,
                                      float nbinv, float w, float& acc) {
    float sm, cm;
    __sincosf(mphi, &sm, &cm);
    float re = wred(P * cm * mk) * nbinv;
    float im = wred(P * sm * mk) * nbinv;
    acc += w * (re * re + im * im);
}

// Accumulate gaussian-of-angle over one 16x16 WMMA C/D tile.
// C/D f32 layout: VGPR r, lanes 0-15 -> M=r, lanes 16-31 -> M=r+8; N=lane&15.
__device__ __forceinline__ float tile_gauss(const v8f D, int ti, int tj,
                                            unsigned mbits, int lane) {
    const int l = (tj << 4) | (lane & 15);
    const int baseM = (ti << 4) + ((lane >> 4) << 3);
    const bool lok = (mbits >> l) & 1u;
    float ssum = 0.0f;
#pragma unroll
    for (int r = 0; r < 8; ++r) {
        const int k = baseM + r;
        const bool kok = (mbits >> k) & 1u;
        float g = fminf(fmaxf(D[r], -1.0f + 1e-7f), 1.0f - 1e-7f);
        float ang = acosf(g);
        float dv = (ang - THETA_T) * INV_SIGMA;
        float gv = __expf(-0.5f * dv * dv);
        ssum += (kok && lok && (k != l)) ? gv : 0.0f;
    }
    return ssum;
}

__global__ __launch_bounds__(256) void order_params_kernel(
    const float* __restrict__ pos, const int* __restrict__ aidx,
    const int* __restrict__ nidx, const int* __restrict__ vmsk,
    float* __restrict__ out, int M, int K) {
    const int lane = threadIdx.x & 31;
    const int atom = blockIdx.x * (blockDim.x >> 5) + (threadIdx.x >> 5);
    if (atom >= M) return;  // wave-uniform: EXEC stays all-1s (WMMA requirement)

    // ---- gather: lane t = neighbor t (t < K), others masked off ----
    const int ai = aidx[atom];
    const float cx = pos[3 * ai + 0];
    const float cy = pos[3 * ai + 1];
    const float cz = pos[3 * ai + 2];

    const int t = lane;
    const int off = atom * K + (t < K ? t : K - 1);
    const int rawn = nidx[off];
    const bool valid = (t < K) && (vmsk[off] > 0);
    const int sidx = valid ? rawn : 0;  // reference: where(mask, idx, 0)
    const float px = pos[3 * sidx + 0];
    const float py = pos[3 * sidx + 1];
    const float pz = pos[3 * sidx + 2];

    float vx = px - cx, vy = py - cy, vz = pz - cz;
    const float dist = sqrtf(vx * vx + vy * vy + vz * vz);
    const float inv = 1.0f / (dist + 1e-10f);
    float ux = vx * inv, uy = vy * inv, uz = vz * inv;
    const float cost = fminf(fmaxf(uz, -1.0f + 1e-7f), 1.0f - 1e-7f);
    const float mk = valid ? 1.0f : 0.0f;
    const float theta = valid ? acosf(cost) : 0.0f;
    const float phi = valid ? atan2f(uy, ux) : 0.0f;
    ux *= mk; uy *= mk; uz *= mk;  // masked units are exactly zero

    const float cn = wred(mk);
    const float nb = fmaxf(cn, 1.0f);
    const float nbinv = 1.0f / nb;
    const unsigned mbits = (unsigned)__ballot(valid);

    // ---- Gram matrix G = U U^T via 4x V_WMMA_F32_16X16X4_F32 ----
    // A fragment (16x4 f32, 2 VGPRs): lanes 0-15 hold (M=lane, K=0/1),
    // lanes 16-31 hold (M=lane-16, K=2/3).  K: 0=x,1=y,2=z,3=zero-pad.
    // B (4x16) uses the mirrored layout, so the same fragment serves both.
    const int half = lane & 15;
    const bool hi = lane >= 16;
    const float sx0 = __shfl(ux, half, 32);
    const float sy0 = __shfl(uy, half, 32);
    const float sz0 = __shfl(uz, half, 32);
    const float sx1 = __shfl(ux, 16 + half, 32);
    const float sy1 = __shfl(uy, 16 + half, 32);
    const float sz1 = __shfl(uz, 16 + half, 32);
    v2f F0, F1;
    F0[0] = hi ? sz0 : sx0;  F0[1] = hi ? 0.0f : sy0;
    F1[0] = hi ? sz1 : sx1;  F1[1] = hi ? 0.0f : sy1;

    const v8f z = {};
    v8f D00 = __builtin_amdgcn_wmma_f32_16x16x4_f32(false, F0, false, F0,
                                                    (short)0, z, false, false);
    v8f D01 = __builtin_amdgcn_wmma_f32_16x16x4_f32(false, F0, false, F1,
                                                    (short)0, z, false, false);
    v8f D10 = __builtin_amdgcn_wmma_f32_16x16x4_f32(false, F1, false, F0,
                                                    (short)0, z, false, false);
    v8f D11 = __builtin_amdgcn_wmma_f32_16x16x4_f32(false, F1, false, F1,
                                                    (short)0, z, false, false);

    float psum = tile_gauss(D00, 0, 0, mbits, lane)
               + tile_gauss(D01, 0, 1, mbits, lane)
               + tile_gauss(D10, 1, 0, mbits, lane)
               + tile_gauss(D11, 1, 1, mbits, lane);
    const float tet = wred(psum) / fmaxf(cn * (cn - 1.0f), 1.0f);

    // ---- Steinhardt q2, q4, q6 ----
    float x, sn;
    __sincosf(theta, &sn, &x);  // theta in [0,pi] -> sn >= 0
    const float x2 = x * x, x3 = x2 * x, x4 = x2 * x2, x5 = x4 * x, x6 = x4 * x2;
    const float s2 = sn * sn, s3 = s2 * sn, s4 = s2 * s2, s5 = s4 * sn, s6 = s4 * s2;

    float acc2 = 0.0f;
    qterm(N20 * 0.5f * (3.0f * x2 - 1.0f), 0.0f,       mk, nbinv, 1.0f, acc2);
    qterm(N21 * (-3.0f * x * sn),          phi,        mk, nbinv, 2.0f, acc2);
    qterm(N22 * (3.0f * s2),               2.0f * phi, mk, nbinv, 2.0f, acc2);
    const float q2 = sqrtf(2.5132741228718345f * acc2);

    float acc4 = 0.0f;
    qterm(N40 * 0.125f * (35.0f * x4 - 30.0f * x2 + 3.0f), 0.0f,       mk, nbinv, 1.0f, acc4);
    qterm(N41 * (-2.5f * (7.0f * x3 - 3.0f * x) * sn),     phi,        mk, nbinv, 2.0f, acc4);
    qterm(N42 * (7.5f * (7.0f * x2 - 1.0f) * s2),          2.0f * phi, mk, nbinv, 2.0f, acc4);
    qterm(N43 * (-105.0f * x * s3),                        3.0f * phi, mk, nbinv, 2.0f, acc4);
    qterm(N44 * (105.0f * s4),                             4.0f * phi, mk, nbinv, 2.0f, acc4);
    const float q4 = sqrtf(1.3962634015954636f * acc4);

    float acc6 = 0.0f;
    qterm(N60 * ((231.0f * x6 - 315.0f * x4 + 105.0f * x2 - 5.0f) * 0.0625f),
          0.0f, mk, nbinv, 1.0f, acc6);
    qterm(N61 * (-2.625f * (33.0f * x5 - 30.0f * x3 + 5.0f * x) * sn),
          phi, mk, nbinv, 2.0f, acc6);
    qterm(N62 * (13.125f * (33.0f * x4 - 18.0f * x2 + 1.0f) * s2),
          2.0f * phi, mk, nbinv, 2.0f, acc6);
    qterm(N63 * (-157.5f * (11.0f * x3 - 3.0f * x) * s3),
          3.0f * phi, mk, nbinv, 2.0f, acc6);
    qterm(N64 * (472.5f * (11.0f * x2 - 1.0f) * s4),
          4.0f * phi, mk, nbinv, 2.0f, acc6);
    qterm(N65 * (-10395.0f * x * s5), 5.0f * phi, mk, nbinv, 2.0f, acc6);
    qterm(N66 * (10395.0f * s6),      6.0f * phi, mk, nbinv, 2.0f, acc6);
    const float q6 = sqrtf(0.9666438934122308f * acc6);

    if (lane == 0) {
        out[atom]         = cn;
        out[M + atom]     = tet;
        out[2 * M + atom] = q2;
        out[3 * M + atom] = q4;
        out[4 * M + atom] = q6;
    }
}

extern "C" void kernel_launch(void* const* d_in, const int* in_sizes, int n_in,
                              void* d_out, int out_size, void* d_ws, size_t ws_size,
                              hipStream_t stream) {
    (void)n_in; (void)out_size; (void)d_ws; (void)ws_size;
    const float* pos  = (const float*)d_in[0];
    const int*   aidx = (const int*)d_in[1];
    const int*   nidx = (const int*)d_in[2];
    const int*   vmsk = (const int*)d_in[3];
    float* out = (float*)d_out;

    const int M = in_sizes[1];
    const int K = in_sizes[2] / M;  // K <= 32 assumed (reference: K = 20)

    const int WPB = 8;  // 8 wave32s per block = 256 threads
    dim3 block(WPB * 32);
    dim3 grid((M + WPB - 1) / WPB);
    hipLaunchKernelGGL(order_params_kernel, grid, block, 0, stream,
                       pos, aidx, nidx, vmsk, out, M, K);
}